// DilateAttention_63883343561390
// MI455X (gfx1250) — compile-verified
//
#include <hip/hip_runtime.h>

// ---------------- problem constants (from reference) ----------------
#define NPOS   8192     // sequence length N
#define DCH    256      // total channels d
#define HEADS  8        // d / HEAD_DIM
#define HD     32       // HEAD_DIM
#define NT     128      // positions per block tile
#define KW     132      // NT + 4 halo (dilation-2, 3-tap window)
#define QSTR   129      // LDS row stride for q tile (pad vs bank conflicts)
#define KSTR   133      // LDS row stride for k/v halo tiles
#define SCALEF 0.17677669529663687f   // 32^-0.5

typedef __attribute__((ext_vector_type(2))) float v2f;
typedef __attribute__((ext_vector_type(8))) float v8f;

// gfx1250 async global->LDS path (ASYNCcnt-tracked), with sync fallback.
// Toolchain declares these builtins with typed int pointers:
//   (int addrspace(1)* gsrc, int addrspace(3)* ldst, imm offset, imm cpol)
#if defined(__AMDGCN__) && __has_builtin(__builtin_amdgcn_global_load_async_to_lds_b32)
#define HAVE_ASYNC_LDS 1
typedef __attribute__((address_space(1))) int as1_int;
typedef __attribute__((address_space(3))) int as3_int;
#else
#define HAVE_ASYNC_LDS 0
#endif

__device__ __forceinline__ void async_copy_f32(const float* gsrc, float* ldst) {
#if HAVE_ASYNC_LDS
    __builtin_amdgcn_global_load_async_to_lds_b32(
        (as1_int*)gsrc, (as3_int*)ldst, 0, 0);
#else
    *ldst = *gsrc;
#endif
}

__device__ __forceinline__ void wait_staging_and_barrier() {
#if HAVE_ASYNC_LDS
#if __has_builtin(__builtin_amdgcn_s_wait_asynccnt)
    __builtin_amdgcn_s_wait_asynccnt(0);
#else
    asm volatile("s_wait_asynccnt 0x0" ::: "memory");
#endif
#endif
    __syncthreads();
}

__device__ __forceinline__ float rdlane(float x, int srcLane) {
    union { float f; int i; } u;
    u.f = x;
    u.i = __builtin_amdgcn_readlane(u.i, srcLane);
    return u.f;
}

__launch_bounds__(256, 2)
__global__ void dilate_attn_wmma_f32(const float* __restrict__ q,
                                     const float* __restrict__ k,
                                     const float* __restrict__ v,
                                     float* __restrict__ out) {
    __shared__ float qs[HD * QSTR];
    __shared__ float ks[HD * KSTR];
    __shared__ float vs[HD * KSTR];

    const int t   = threadIdx.x;
    const int n0  = blockIdx.x * NT;           // tile start along sequence
    const int h   = blockIdx.y;                // head
    const int b   = blockIdx.z;                // batch
    const int cb  = b * DCH + h * HD;          // global channel base for this head

    // ---- stage Q tile: qs[d][ln], coalesced, direct global->LDS ----
    for (int i = t; i < HD * NT; i += 256) {
        const int d  = i >> 7;          // /128
        const int ln = i & (NT - 1);
        async_copy_f32(&q[(size_t)(cb + d) * NPOS + n0 + ln], &qs[d * QSTR + ln]);
    }
    // ---- stage K/V halo tiles: [d][j], j covers n0-2 .. n0+129, zero padded ----
    for (int i = t; i < HD * KW; i += 256) {
        const int d = i / KW;
        const int j = i - d * KW;
        const int n = n0 - 2 + j;
        if ((unsigned)n < (unsigned)NPOS) {
            const size_t gi = (size_t)(cb + d) * NPOS + n;
            async_copy_f32(&k[gi], &ks[d * KSTR + j]);
            async_copy_f32(&v[gi], &vs[d * KSTR + j]);
        } else {
            // edge blocks only: zero-pad halo (written via DS pipe; disjoint
            // from any async destination, ordered by the barrier below)
            ks[d * KSTR + j] = 0.0f;
            vs[d * KSTR + j] = 0.0f;
        }
    }
    wait_staging_and_barrier();

    const int lane = t & 31;
    const int wave = t >> 5;          // 0..7  -> each wave owns 16 positions
    const int tile = wave * 16;       // local tile start within the 128 block
    const int mr   = lane & 15;       // A row (M) / B column (N) index
    const int g    = lane >> 4;       // lane half selects K sub-pair

    // A fragments (Q), reused across the 3 shifts.
    // 32-bit A 16x4 layout: lane half g, VGPR p  ->  K = 2*g + p; per step kk,
    // matrix K (channel d) = 4*kk + 2*g + p.
    v2f afrag[8];
#pragma unroll
    for (int kk = 0; kk < 8; ++kk) {
        const int dbase = 4 * kk + 2 * g;
        afrag[kk].x = qs[(dbase + 0) * QSTR + tile + mr];
        afrag[kk].y = qs[(dbase + 1) * QSTR + tile + mr];
    }

    // Scores for the 3 dilated taps: C[s][m][n] = <q_{tile+m}, k_{tile+n+2s-2}>
    // (only the diagonal m==n is used).
    v8f c[3];
#pragma unroll
    for (int s = 0; s < 3; ++s) {
        v8f acc = {};
        const int jcol = tile + mr + 2 * s;   // halo-space column for B
#pragma unroll
        for (int kk = 0; kk < 8; ++kk) {
            const int dbase = 4 * kk + 2 * g;
            v2f bfrag;
            bfrag.x = ks[(dbase + 0) * KSTR + jcol];
            bfrag.y = ks[(dbase + 1) * KSTR + jcol];
            acc = __builtin_amdgcn_wmma_f32_16x16x4_f32(
                      /*neg_a=*/false, afrag[kk],
                      /*neg_b=*/false, bfrag,
                      /*c_mod=*/(short)0, acc,
                      /*reuse_a=*/false, /*reuse_b=*/false);
        }
        c[s] = acc;
    }

    // ---- softmax (incl. the 6 zero-logit padding slots) + V blend ----
    // C layout: lane L: N = L%16 ; VGPR r: M = 8*(L/16) + r.
    // Diagonal m: m<8 -> (lane m, comp m); m>=8 -> (lane m+16, comp m-8).
    const size_t outBase = ((size_t)b * NPOS) * DCH + (size_t)h * HD + lane;
#pragma unroll
    for (int m = 0; m < 16; ++m) {
        const int srcLane = (m < 8) ? m : (m + 16);
        const int comp    = (m < 8) ? m : (m - 8);
        const float s0 = SCALEF * rdlane(c[0][comp], srcLane);
        const float s1 = SCALEF * rdlane(c[1][comp], srcLane);
        const float s2 = SCALEF * rdlane(c[2][comp], srcLane);
        const float mx = fmaxf(fmaxf(s0, s1), fmaxf(s2, 0.0f));
        const float e0 = __expf(s0 - mx);
        const float e1 = __expf(s1 - mx);
        const float e2 = __expf(s2 - mx);
        const float inv = 1.0f / (6.0f * __expf(-mx) + e0 + e1 + e2);
        const int jj = tile + m;   // halo-space index of tap 0 for this position
        const float o = (e0 * vs[lane * KSTR + jj] +
                         e1 * vs[lane * KSTR + jj + 2] +
                         e2 * vs[lane * KSTR + jj + 4]) * inv;
        const int n = n0 + tile + m;
        out[outBase + (size_t)n * DCH] = o;   // 128B coalesced per wave
    }
}

extern "C" void kernel_launch(void* const* d_in, const int* in_sizes, int n_in,
                              void* d_out, int out_size, void* d_ws, size_t ws_size,
                              hipStream_t stream) {
    const float* q = (const float*)d_in[0];
    const float* k = (const float*)d_in[1];
    const float* v = (const float*)d_in[2];
    float* out = (float*)d_out;

    const int B = in_sizes[0] / (DCH * NPOS);   // 4 per reference
    dim3 grid(NPOS / NT, HEADS, B);
    dim3 block(256, 1, 1);
    dilate_attn_wmma_f32<<<grid, block, 0, stream>>>(q, k, v, out);
}